// Projection_5171140624940
// MI455X (gfx1250) — compile-verified
//
#include <hip/hip_runtime.h>
#include <hip/hip_bf16.h>

// Problem constants (fixed by the reference).
#define BATCH      524288
#define TOTAL_K    65          // sum(FEATURE_DIMS)
#define TOTAL_N    230         // F * EMBEDDING_DIM
#define NTILES     15          // ceil(240/16) padded N tiles
#define CROWS      112         // compact Wbig rows staged in LDS (= 4 * sum(steps))

typedef __attribute__((ext_vector_type(2))) float v2f;
typedef __attribute__((ext_vector_type(8))) float v8f;

// Per-N-tile K window start, number of 4-wide K steps, and compact-row base.
// Derived from FEATURE_DIMS = [4x12, 2,2,1,4,1,1,1,1,1,1,2] (offsets below).
constexpr int KS[NTILES]    = {0, 4, 12, 16, 24, 32, 36, 44, 48, 52, 57, 58, 60, 61, 63};
constexpr int STEPS[NTILES] = {2, 3,  2,  3,  2,  2,  3,  2,  2,  2,  1,  1,  1,  1,  1};
constexpr int RB[NTILES + 1] = {0, 8, 20, 28, 40, 48, 56, 68, 76, 84, 92, 96, 100, 104, 108, 112};

// Feature offsets/dims (runtime-indexed -> constant memory).
__constant__ int c_off[23] = {0,4,8,12,16,20,24,28,32,36,40,44,48,50,52,53,57,58,59,60,61,62,63};
__constant__ int c_dim[23] = {4,4,4,4,4,4,4,4,4,4,4,4, 2, 2, 1, 4, 1, 1, 1, 1, 1, 1, 2};

__global__ __launch_bounds__(256) void projection_wmma_kernel(
    const float* __restrict__ x,      // (B, 65)
    const float* __restrict__ W,      // (23, 4, 10), padded rows already zero
    const float* __restrict__ bias,   // (23, 10) == flat (230,)
    float* __restrict__ out)          // (B, 230)
{
    // Compact block-diagonal weight tiles: 112 rows x 16 cols (7 KB) + padded bias.
    __shared__ float sW[CROWS * 16];
    __shared__ float sB[240];

    const int tid = threadIdx.x;

    // ---- Stage compact Wbig into LDS (1792 cells / 256 threads = 7 each) ----
#pragma unroll
    for (int it = 0; it < 7; ++it) {
        const int cell = tid + it * 256;
        const int r  = cell >> 4;     // compact row 0..111
        const int cc = cell & 15;     // column within tile
        int tile = 0, k = 0;
#pragma unroll
        for (int t = 0; t < NTILES; ++t) {
            if (r >= RB[t]) { tile = t; k = KS[t] + (r - RB[t]); }  // global K row
        }
        const int gcol = tile * 16 + cc;                            // global N col
        float val = 0.0f;
        if (gcol < TOTAL_N) {
            const int f = gcol / 10;
            const int e = gcol - f * 10;
            const int d = k - c_off[f];
            if (d >= 0 && d < c_dim[f]) val = W[(f * 4 + d) * 10 + e];
        }
        sW[cell] = val;
    }
    if (tid < 240) sB[tid] = (tid < TOTAL_N) ? bias[tid] : 0.0f;
    __syncthreads();

    // ---- Per-wave 16-row M tile ----
    const int lane = tid & 31;
    const int wave = tid >> 5;
    const int h    = lane >> 4;   // lane half: K phase for A/B operands
    const int nl   = lane & 15;   // M index (A) / N index (B, C)
    const int m0   = (blockIdx.x * 8 + wave) * 16;
    const int xrow = (m0 + nl) * TOTAL_K;   // contiguous 16x65 slice per wave

#pragma unroll
    for (int n = 0; n < NTILES; ++n) {
        const int ks = KS[n];
        const int st = STEPS[n];
        const int rb = RB[n];
        const int gcol = n * 16 + nl;

        // C init = bias[col]; every C VGPR of this lane maps to the same column.
        const float bv = sB[n * 16 + nl];
        v8f c = {bv, bv, bv, bv, bv, bv, bv, bv};

#pragma unroll
        for (int s = 0; s < st; ++s) {
            const int kA = ks + 4 * s + 2 * h;   // K for VGPR0; VGPR1 is kA+1
            v2f a;
            a.x = (kA     < TOTAL_K) ? x[xrow + kA]     : 0.0f;  // folds away except tile 14
            a.y = (kA + 1 < TOTAL_K) ? x[xrow + kA + 1] : 0.0f;
            v2f b;
            const int lb = (rb + 4 * s + 2 * h) * 16 + nl;
            b.x = sW[lb];
            b.y = sW[lb + 16];
            // D = A(16x4) * B(4x16) + C ; f32 WMMA, wave32.
            c = __builtin_amdgcn_wmma_f32_16x16x4_f32(
                    false, a, false, b, (short)0, c, false, false);
        }

        if (gcol < TOTAL_N) {                    // only tile 14 is partial
#pragma unroll
            for (int r = 0; r < 8; ++r) {
                // C layout: VGPR r, lane half h -> row m0 + r + 8h, col gcol.
                out[(m0 + r + 8 * h) * TOTAL_N + gcol] = c[r];
            }
        }
    }
}

extern "C" void kernel_launch(void* const* d_in, const int* in_sizes, int n_in,
                              void* d_out, int out_size, void* d_ws, size_t ws_size,
                              hipStream_t stream) {
    const float* x    = (const float*)d_in[0];   // input (B, 65) f32
    const float* W    = (const float*)d_in[1];   // (23, 4, 10) f32
    const float* bias = (const float*)d_in[2];   // (23, 10) f32
    // d_in[3] (idx) unused: FEATURE_DIMS are compile-time constants.
    float* out = (float*)d_out;

    const int B = in_sizes[0] / TOTAL_K;         // 524288
    const int grid = B / (16 * 8);               // one 16-row tile per wave, 8 waves/block
    projection_wmma_kernel<<<grid, 256, 0, stream>>>(x, W, bias, out);
}